// ChamferDistanceL2_withnormal_55482387530101
// MI455X (gfx1250) — compile-verified
//
#include <hip/hip_runtime.h>

typedef __attribute__((ext_vector_type(2))) float v2f;
typedef __attribute__((ext_vector_type(8))) float v8f;

#define NB 8
#define NP 4096
#define ROWS_PER_WG 128   // 8 waves * 16 rows each
#define WG_THREADS 256
#define COL_TILES (NP / 16)

__device__ __forceinline__ unsigned long long umin64(unsigned long long a, unsigned long long b) {
    return a < b ? a : b;   // lowers to v_min_u64
}
__device__ __forceinline__ unsigned long long shfl_xor_u64(unsigned long long v, int m) {
    int lo = __shfl_xor((int)(unsigned int)v, m, 32);
    int hi = __shfl_xor((int)(unsigned int)(v >> 32), m, 32);
    return ((unsigned long long)(unsigned int)hi << 32) | (unsigned int)lo;
}
__device__ __forceinline__ unsigned long long pack64(float d, unsigned int idx) {
    // d >= 0 in exact arithmetic, so raw IEEE bits compare correctly as unsigned.
    // (A tiny negative rounding artifact would sort last instead of first --
    //  negligible for random point clouds, and it saves 2 VALU ops/element.)
    return (((unsigned long long)__float_as_uint(d)) << 32) | idx;
}

// ---------------- init: best arrays to +inf-pattern, out to 0 ----------------
__global__ void chamfer_init_kernel(unsigned long long* ws64, float* out) {
    int i = blockIdx.x * blockDim.x + threadIdx.x;
    if (i < 2 * NB * NP) ws64[i] = ~0ULL;
    if (i == 0) out[0] = 0.0f;
}

// ---- one 16x16 distance tile:  D[m][n] = C[m] + A[m,:].B[:,n] ----
//   A[m][0..2] = -2*p1[m],  A[m][3] = 1
//   B[0..2][n] =  p2[n],    B[3][n] = |p2[n]|^2
//   C[m][n]    = |p1[m]|^2          (loop-invariant, SRC2 != VDST -> no copies)
__device__ __forceinline__ void process_tile(
    int colBase, int lo, int hi, int rowLaneBase,
    float x2, float y2, float z2,
    const v2f& a, const v8f& cinit,
    unsigned long long rmin[8],
    unsigned long long* colLocal)
{
    const unsigned int col = (unsigned int)(colBase + lo);
    const float sq2 = x2 * x2 + y2 * y2 + z2 * z2;

    v2f bb;
    if (hi == 0) { bb.x = x2; bb.y = y2;  }
    else         { bb.x = z2; bb.y = sq2; }

    const v8f d = __builtin_amdgcn_wmma_f32_16x16x4_f32(
        false, a, false, bb, (short)0, cinit, false, false);

    unsigned long long cp[8];
#pragma unroll
    for (int v = 0; v < 8; ++v) {
        rmin[v] = umin64(rmin[v], pack64(d[v], col));                 // row-min: idx = col
        cp[v]   = pack64(d[v], (unsigned int)(rowLaneBase + v));      // col-min: idx = row
    }
    const unsigned long long m0 = umin64(umin64(cp[0], cp[1]), umin64(cp[2], cp[3]));
    const unsigned long long m1 = umin64(umin64(cp[4], cp[5]), umin64(cp[6], cp[7]));
    atomicMin(&colLocal[col], umin64(m0, m1));   // all 32 lanes; halves cover different rows
}

// ---------------- distance matrix + min/argmin via WMMA ----------------
// grid: NB * (NP/ROWS_PER_WG) blocks of 256 threads (8 waves).
// wave w owns rows [rowBlk*128 + w*16, +16), sweeps all 256 column tiles.
__global__ __launch_bounds__(WG_THREADS) void chamfer_dist_kernel(
    const float* __restrict__ xyz1, const float* __restrict__ xyz2,
    unsigned long long* __restrict__ rowBest, unsigned long long* __restrict__ colBest)
{
    __shared__ unsigned long long colLocal[NP];   // 32 KB
    const int tid = threadIdx.x;
    for (int i = tid; i < NP; i += WG_THREADS) colLocal[i] = ~0ULL;
    __syncthreads();

    const int b      = blockIdx.x / (NP / ROWS_PER_WG);
    const int rowBlk = blockIdx.x % (NP / ROWS_PER_WG);
    const int wave   = tid >> 5;
    const int lane   = tid & 31;
    const int lo     = lane & 15;
    const int hi     = lane >> 4;

    const float* p1base = xyz1 + (size_t)b * NP * 6;
    const float* p2base = xyz2 + (size_t)b * NP * 6;

    // ---- A fragment + C fragment (fixed for the whole column sweep) ----
    const int rowTileBase = rowBlk * ROWS_PER_WG + wave * 16;
    const int myRow = rowTileBase + lo;
    const float* pr = p1base + (size_t)myRow * 6;
    const float x1 = pr[0], y1 = pr[1], z1 = pr[2];
    const float sq1 = x1 * x1 + y1 * y1 + z1 * z1;

    v2f a;
    if (hi == 0) { a.x = -2.0f * x1; a.y = -2.0f * y1; }
    else         { a.x = -2.0f * z1; a.y = 1.0f;       }

    // C[m][n] = |p1[m]|^2 : lane needs rows (v + 8*hi), held by lanes 0..15
    v8f cinit;
#pragma unroll
    for (int v = 0; v < 8; ++v) cinit[v] = __shfl(sq1, v + 8 * hi, 32);

    const int rowLaneBase = rowTileBase + 8 * hi;   // + v = global row of D[v]

    unsigned long long rmin[8];
#pragma unroll
    for (int v = 0; v < 8; ++v) rmin[v] = ~0ULL;

    // ---- software-pipelined, 2x-unrolled column sweep ----
    const float* p2l = p2base + (size_t)lo * 6;      // lane's slot in tile 0
    float xa = p2l[0],  ya = p2l[1],  za = p2l[2];   // tile 0
    float xb = p2l[96], yb = p2l[97], zb = p2l[98];  // tile 1 (16 cols * 6 floats)

    for (int ct = 0; ct < COL_TILES; ct += 2) {
        // issue the next two tiles' B loads early (wrap keeps them in-bounds)
        const int n0 = ((ct + 2) & (COL_TILES - 1)) * 96;
        const int n1 = ((ct + 3) & (COL_TILES - 1)) * 96;
        const float nxa = p2l[n0], nya = p2l[n0 + 1], nza = p2l[n0 + 2];
        const float nxb = p2l[n1], nyb = p2l[n1 + 1], nzb = p2l[n1 + 2];

        process_tile(ct * 16,      lo, hi, rowLaneBase, xa, ya, za, a, cinit, rmin, colLocal);
        process_tile(ct * 16 + 16, lo, hi, rowLaneBase, xb, yb, zb, a, cinit, rmin, colLocal);

        xa = nxa; ya = nya; za = nza;
        xb = nxb; yb = nyb; zb = nzb;
    }

    // reduce row-mins across the 16 lanes of each half (columns mod 16)
#pragma unroll
    for (int v = 0; v < 8; ++v) {
        unsigned long long r = rmin[v];
#pragma unroll
        for (int m = 1; m <= 8; m <<= 1) r = umin64(r, shfl_xor_u64(r, m));
        rmin[v] = r;
    }
    if (lo == 0) {  // lane 0 -> rows 0..7 of tile, lane 16 -> rows 8..15
#pragma unroll
        for (int v = 0; v < 8; ++v)
            rowBest[(size_t)b * NP + rowLaneBase + v] = rmin[v];
    }

    __syncthreads();
    for (int i = tid; i < NP; i += WG_THREADS)
        atomicMin(&colBest[(size_t)b * NP + i], colLocal[i]);
}

// ---------------- gather normals, normalize, final reduction ----------------
__device__ __forceinline__ float norm_diff2(const float* a, const float* b) {
    const float ax = a[0], ay = a[1], az = a[2];
    const float bx = b[0], by = b[1], bz = b[2];
    const float na = sqrtf(ax * ax + ay * ay + az * az);
    const float nb = sqrtf(bx * bx + by * by + bz * bz);
    const float ia = 1.0f / fmaxf(na, 1e-12f);
    const float ib = 1.0f / fmaxf(nb, 1e-12f);
    const float dx = ax * ia - bx * ib;
    const float dy = ay * ia - by * ib;
    const float dz = az * ia - bz * ib;
    return dx * dx + dy * dy + dz * dz;
}

__global__ __launch_bounds__(256) void chamfer_reduce_kernel(
    const float* __restrict__ xyz1, const float* __restrict__ xyz2,
    const unsigned long long* __restrict__ rowBest,
    const unsigned long long* __restrict__ colBest,
    float* __restrict__ out)
{
    __shared__ float red[256];
    const int i = blockIdx.x * 256 + threadIdx.x;   // 0 .. NB*NP-1
    const int b = i >> 12;
    const int n = i & (NP - 1);

    const unsigned long long k1 = rowBest[i];
    const float dist1 = __uint_as_float((unsigned int)(k1 >> 32));
    const int   idx1  = (int)(unsigned int)(k1 & 0xFFFFFFFFu);

    const unsigned long long k2 = colBest[i];
    const float dist2 = __uint_as_float((unsigned int)(k2 >> 32));
    const int   idx2  = (int)(unsigned int)(k2 & 0xFFFFFFFFu);

    const float* r1 = xyz1 + ((size_t)b * NP + n) * 6;     // point n in cloud 1
    const float* r2 = xyz2 + ((size_t)b * NP + n) * 6;     // point n in cloud 2
    const float* t1 = xyz2 + ((size_t)b * NP + idx1) * 6;  // nearest in cloud 2
    const float* t2 = xyz1 + ((size_t)b * NP + idx2) * 6;  // nearest in cloud 1

    const float nd1 = norm_diff2(r1 + 3, t1 + 3);
    const float nd2 = norm_diff2(r2 + 3, t2 + 3);

    const float val = (dist1 + dist2 + nd1 + nd2) * (1.0f / (float)(NB * NP));

    red[threadIdx.x] = val;
    __syncthreads();
    for (int s = 128; s > 0; s >>= 1) {
        if (threadIdx.x < s) red[threadIdx.x] += red[threadIdx.x + s];
        __syncthreads();
    }
    if (threadIdx.x == 0) atomicAdd(out, red[0]);
}

extern "C" void kernel_launch(void* const* d_in, const int* in_sizes, int n_in,
                              void* d_out, int out_size, void* d_ws, size_t ws_size,
                              hipStream_t stream) {
    const float* xyz1 = (const float*)d_in[0];
    const float* xyz2 = (const float*)d_in[1];
    float* out = (float*)d_out;

    unsigned long long* rowBest = (unsigned long long*)d_ws;       // NB*NP u64
    unsigned long long* colBest = rowBest + (size_t)NB * NP;       // NB*NP u64

    chamfer_init_kernel<<<(2 * NB * NP + 255) / 256, 256, 0, stream>>>(rowBest, out);

    chamfer_dist_kernel<<<NB * (NP / ROWS_PER_WG), WG_THREADS, 0, stream>>>(
        xyz1, xyz2, rowBest, colBest);

    chamfer_reduce_kernel<<<(NB * NP) / 256, 256, 0, stream>>>(
        xyz1, xyz2, rowBest, colBest, out);
}